// GRPNet_63230508532370
// MI455X (gfx1250) — compile-verified
//
#include <hip/hip_runtime.h>
#include <hip/hip_bf16.h>
#include <math.h>

typedef float v2f __attribute__((ext_vector_type(2)));
typedef float v8f __attribute__((ext_vector_type(8)));

#define D_  64
#define A_  8
#define H_  256
#define N_  16
#define T_  128
#define E_  512

// hyperparams: L2=SF2=SN2=0.01 ;  k(d) = SF2*exp(-0.5*d^2/L2) = 0.01*exp(-50*d^2)
__device__ __forceinline__ float rbf(float d) {
    return 0.01f * expf(-50.0f * d * d);
}

// ---------------- Kernel 1: fused MLP -> times (E,N), anchors (E, A*N) --------
__global__ __launch_bounds__(256) void mlp_kernel(
    const float* __restrict__ x,
    const float* __restrict__ W1, const float* __restrict__ b1,
    const float* __restrict__ W2, const float* __restrict__ b2,
    const float* __restrict__ Wt, const float* __restrict__ bt,
    const float* __restrict__ Wa, const float* __restrict__ ba,
    float* __restrict__ times, float* __restrict__ anchors)
{
    __shared__ float xs[D_];
    __shared__ float h1[H_];
    __shared__ float h2[H_];
    const int e   = blockIdx.x;
    const int tid = threadIdx.x;

    if (tid < D_) xs[tid] = x[e * D_ + tid];
    __syncthreads();
    {
        float s = b1[tid];
        const float* w = W1 + tid * D_;
        #pragma unroll 8
        for (int k = 0; k < D_; ++k) s += w[k] * xs[k];
        h1[tid] = tanhf(s);
    }
    __syncthreads();
    {
        float s = b2[tid];
        const float* w = W2 + tid * H_;
        #pragma unroll 8
        for (int k = 0; k < H_; ++k) s += w[k] * h1[k];
        h2[tid] = tanhf(s);
    }
    __syncthreads();
    if (tid < N_) {
        float s = bt[tid];
        const float* w = Wt + tid * H_;
        #pragma unroll 8
        for (int k = 0; k < H_; ++k) s += w[k] * h2[k];
        times[e * N_ + tid] = s;
    }
    if (tid < N_ * A_) {
        float s = ba[tid];
        const float* w = Wa + tid * H_;
        #pragma unroll 8
        for (int k = 0; k < H_; ++k) s += w[k] * h2[k];
        anchors[e * (N_ * A_) + tid] = s;   // layout: e*128 + a*16 + n
    }
}

// ---------------- Kernel 2: Kss (T x T), shared by ALL problems ---------------
__global__ void kss_kernel(const float* __restrict__ tq, float* __restrict__ kss)
{
    const int i = blockIdx.x, j = threadIdx.x;
    const float d = tq[i] - tq[j];
    kss[i * T_ + j] = rbf(d);
}

// ---------------- Kernel 3: one block per sample e ----------------------------
// 128 threads = 4 wave32. Cholesky + triangular solves in LDS, cov via
// v_wmma_f32_16x16x4_f32, dedup across the 8 anchors (cov is a-independent).
__global__ __launch_bounds__(128) void gp_kernel(
    const float* __restrict__ times, const float* __restrict__ anchors,
    const float* __restrict__ tq_g, const float* __restrict__ kss_g,
    float* __restrict__ mu_out, float* __restrict__ cov_out)
{
    __shared__ float tqs[T_];
    __shared__ float ts[N_];
    __shared__ float Kc[N_ * 17];      // K, then in-place Cholesky L (lower)
    __shared__ float Ks[T_ * 17];      // (T x N), row stride 17 (bank-conflict pad)
    __shared__ float Vs[N_ * T_];      // V = K^{-1} Ks^T  (N x T)
    __shared__ float alph[A_ * N_];    // K^{-1} y per anchor

    const int e     = blockIdx.x;
    const int tid   = threadIdx.x;
    const int lane  = tid & 31;
    const int wave  = tid >> 5;
    const int lrow  = lane & 15;   // row/col index within half-wave
    const int khalf = lane >> 4;   // 0: K pair {0,1}; 1: K pair {2,3}

    tqs[tid] = tq_g[tid];
    if (tid < N_) ts[tid] = times[e * N_ + tid];
    __syncthreads();

    // K = rbf(t,t) + SN2*I
    for (int idx = tid; idx < N_ * N_; idx += 128) {
        const int i = idx >> 4, j = idx & 15;
        float v = rbf(ts[i] - ts[j]);
        if (i == j) v += 0.01f;                     // SN2
        Kc[i * 17 + j] = v;
    }
    // Ks = rbf(tq, t)   (T x N)
    for (int idx = tid; idx < T_ * N_; idx += 128) {
        const int i = idx >> 4, j = idx & 15;
        Ks[i * 17 + j] = rbf(tqs[i] - ts[j]);
    }
    __syncthreads();

    // In-place Cholesky of Kc (lower triangle), column-sequential, row-parallel.
    for (int j = 0; j < N_; ++j) {
        if (tid == 0) {
            float s = Kc[j * 17 + j];
            for (int p = 0; p < j; ++p) { const float l = Kc[j * 17 + p]; s -= l * l; }
            Kc[j * 17 + j] = sqrtf(s);
        }
        __syncthreads();
        if (tid > j && tid < N_) {
            float s = Kc[tid * 17 + j];
            for (int p = 0; p < j; ++p) s -= Kc[tid * 17 + p] * Kc[j * 17 + p];
            Kc[tid * 17 + j] = s / Kc[j * 17 + j];
        }
        __syncthreads();
    }

    // V = K^{-1} Ks^T : one RHS column per thread (RHS col = row of Ks)
    {
        float z[N_];
        const int col = tid;
        #pragma unroll
        for (int i = 0; i < N_; ++i) {             // forward:  L z = Ks[col,:]
            float s = Ks[col * 17 + i];
            for (int p = 0; p < i; ++p) s -= Kc[i * 17 + p] * z[p];
            z[i] = s / Kc[i * 17 + i];
        }
        #pragma unroll
        for (int i = N_ - 1; i >= 0; --i) {        // backward: L^T v = z (in place)
            float s = z[i];
            for (int p = i + 1; p < N_; ++p) s -= Kc[p * 17 + i] * z[p];
            z[i] = s / Kc[i * 17 + i];
        }
        #pragma unroll
        for (int i = 0; i < N_; ++i) Vs[i * T_ + col] = z[i];
    }
    // alpha = K^{-1} y, one anchor per thread (a-dependent part)
    if (tid < A_) {
        float z[N_];
        const float* y = anchors + (size_t)e * (A_ * N_) + tid * N_;
        #pragma unroll
        for (int i = 0; i < N_; ++i) {
            float s = y[i];
            for (int p = 0; p < i; ++p) s -= Kc[i * 17 + p] * z[p];
            z[i] = s / Kc[i * 17 + i];
        }
        #pragma unroll
        for (int i = N_ - 1; i >= 0; --i) {
            float s = z[i];
            for (int p = i + 1; p < N_; ++p) s -= Kc[p * 17 + i] * z[p];
            z[i] = s / Kc[i * 17 + i];
        }
        #pragma unroll
        for (int i = 0; i < N_; ++i) alph[tid * N_ + i] = z[i];
    }
    __syncthreads();

    // mu[a,i] = Ks[i,:] . alpha[a,:]
    for (int idx = tid; idx < A_ * T_; idx += 128) {
        const int a = idx >> 7, i = idx & 127;
        float s = 0.0f;
        #pragma unroll
        for (int j = 0; j < N_; ++j) s += Ks[i * 17 + j] * alph[a * N_ + j];
        __builtin_nontemporal_store(s, mu_out + ((size_t)(e * A_ + a)) * T_ + i);
    }

    // cov = Kss - Ks @ V : 8x8 grid of 16x16 tiles, 4 x WMMA_F32_16X16X4_F32 each.
    // f32 A 16x4 layout: lanes 0-15 hold (K=0,K=1) of row M=lane; lanes 16-31 (K=2,K=3).
    // f32 B 4x16 layout: lanes 0-15 hold rows K=0,K=1 at N=lane; lanes 16-31 rows K=2,K=3.
    for (int tile = wave; tile < 64; tile += 4) {
        const int m0 = (tile >> 3) << 4;
        const int n0 = (tile & 7) << 4;
        v8f c = {0.f, 0.f, 0.f, 0.f, 0.f, 0.f, 0.f, 0.f};
        #pragma unroll
        for (int kc = 0; kc < 4; ++kc) {
            const int k0 = kc * 4 + khalf * 2;
            v2f a_frag, b_frag;
            a_frag.x = Ks[(m0 + lrow) * 17 + k0 + 0];
            a_frag.y = Ks[(m0 + lrow) * 17 + k0 + 1];
            b_frag.x = Vs[(k0 + 0) * T_ + n0 + lrow];
            b_frag.y = Vs[(k0 + 1) * T_ + n0 + lrow];
            c = __builtin_amdgcn_wmma_f32_16x16x4_f32(
                    false, a_frag, false, b_frag, (short)0, c, false, false);
        }
        const size_t ebase = (size_t)e * A_ * (T_ * T_);
        #pragma unroll
        for (int r = 0; r < 8; ++r) {
            const int row = m0 + r + khalf * 8;     // C/D: vgpr r -> M=r (+8 for upper half)
            const int col = n0 + lrow;
            const float val = kss_g[row * T_ + col] - c[r];
            float* p = cov_out + ebase + (size_t)row * T_ + col;
            #pragma unroll
            for (int a = 0; a < A_; ++a)            // cov identical across anchors
                __builtin_nontemporal_store(val, p + (size_t)a * (T_ * T_));
        }
    }
}

extern "C" void kernel_launch(void* const* d_in, const int* in_sizes, int n_in,
                              void* d_out, int out_size, void* d_ws, size_t ws_size,
                              hipStream_t stream)
{
    (void)in_sizes; (void)n_in; (void)out_size; (void)ws_size;
    const float* x  = (const float*)d_in[0];
    // d_in[1] = a, d_in[2] = da : unused by the reference
    const float* W1 = (const float*)d_in[3];
    const float* b1 = (const float*)d_in[4];
    const float* W2 = (const float*)d_in[5];
    const float* b2 = (const float*)d_in[6];
    const float* Wt = (const float*)d_in[7];
    const float* bt = (const float*)d_in[8];
    const float* Wa = (const float*)d_in[9];
    const float* ba = (const float*)d_in[10];
    const float* tq = (const float*)d_in[11];

    float* out = (float*)d_out;
    float* ws  = (float*)d_ws;

    float* times   = ws;                               // 512*16
    float* anchors = ws + E_ * N_;                     // 512*128
    float* kss     = ws + E_ * N_ + E_ * A_ * N_;      // 128*128

    float* mu_out  = out;                              // (E,A,T)
    float* cov_out = out + (size_t)E_ * A_ * T_;       // (E,A,T,T)

    mlp_kernel<<<E_, 256, 0, stream>>>(x, W1, b1, W2, b2, Wt, bt, Wa, ba, times, anchors);
    kss_kernel<<<T_, T_, 0, stream>>>(tq, kss);
    gp_kernel<<<E_, 128, 0, stream>>>(times, anchors, tq, kss, mu_out, cov_out);
}